// CoordMP_25958782337714
// MI455X (gfx1250) — compile-verified
//
#include <hip/hip_runtime.h>
#include <hip/hip_bf16.h>
#include <math.h>

// N=1024, D=64 fixed by the reference.
#define NPTS 1024
#define DDIM 64

typedef __attribute__((ext_vector_type(2))) float v2f;
typedef __attribute__((ext_vector_type(8))) float v8f;

// ---------------------------------------------------------------------------
// Generic (M=1024, K=64, N=64) f32 GEMM: out = A @ B (+ bias per column).
// One wave computes one 16x16 output tile; 4 waves/block cover the 4 N-tiles,
// grid.x = 64 covers the M-tiles. K=64 -> 16 chained V_WMMA_F32_16X16X4_F32.
//
// ISA lane layouts (CDNA5 ISA):
//   A 16x4 : lanes 0-15 hold M=lane,   v[0]=K0, v[1]=K1; lanes 16-31: K2,K3
//   B 4x16 : lanes 0-15 hold N=lane,   v[0]=K0, v[1]=K1; lanes 16-31: K2,K3
//   C 16x16: VGPR r, lanes 0-15 -> M=r, lanes 16-31 -> M=r+8; N = lane&15
// ---------------------------------------------------------------------------
__global__ void wmma_gemm_1024x64x64(const float* __restrict__ A,
                                     const float* __restrict__ B,
                                     const float* __restrict__ bias,
                                     float* __restrict__ out) {
    const int lane = threadIdx.x & 31;
    const int wave = threadIdx.x >> 5;      // N-tile 0..3
    const int m0   = blockIdx.x * 16;
    const int n0   = wave * 16;
    const int half = lane >> 4;             // 0: K pair {0,1}, 1: K pair {2,3}
    const int l15  = lane & 15;
    const int arow = m0 + l15;
    const int bcol = n0 + l15;

    v8f c = {};
#pragma unroll
    for (int kk = 0; kk < DDIM; kk += 4) {
        const int ka = kk + 2 * half;
        v2f a, b;
        a[0] = A[arow * DDIM + ka + 0];
        a[1] = A[arow * DDIM + ka + 1];
        b[0] = B[(ka + 0) * DDIM + bcol];
        b[1] = B[(ka + 1) * DDIM + bcol];
        c = __builtin_amdgcn_wmma_f32_16x16x4_f32(
                /*neg_a=*/false, a, /*neg_b=*/false, b,
                /*c_mod=*/(short)0, c, /*reuse_a=*/false, /*reuse_b=*/false);
    }

    const float bv = bias ? bias[bcol] : 0.0f;
    const int orow0 = m0 + half * 8;
#pragma unroll
    for (int r = 0; r < 8; ++r) {
        out[(orow0 + r) * DDIM + bcol] = c[r] + bv;
    }
}

// ---------------------------------------------------------------------------
// cc[i] = x[i,:] . Wc[0:64]   + bc      (gate "i" term, bias folded)
// dd[i] = x[i,:] . Wc[64:128]           (gate "j" term)
// ---------------------------------------------------------------------------
__global__ void gemv_wc(const float* __restrict__ x,
                        const float* __restrict__ Wc,
                        const float* __restrict__ bc,
                        float* __restrict__ cc,
                        float* __restrict__ dd) {
    const int i = blockIdx.x * blockDim.x + threadIdx.x;
    if (i >= NPTS) return;
    float s0 = 0.f, s1 = 0.f;
#pragma unroll 8
    for (int k = 0; k < DDIM; ++k) {
        const float xv = x[i * DDIM + k];
        s0 += xv * Wc[k];
        s1 += xv * Wc[DDIM + k];
    }
    cc[i] = s0 + bc[0];
    dd[i] = s1;
}

__device__ __forceinline__ float silu_fast(float h) {
    // h * sigmoid(h); fast exp + hardware reciprocal (v_exp_f32 / v_rcp_f32)
    return h * __builtin_amdgcn_rcpf(1.0f + __expf(-h));
}

// ---------------------------------------------------------------------------
// Pair loop. One block per i (8 waves x 32 lanes). Each wave strides over j.
// Lane d accumulates silu(h) for feature dims d and d+32 (coalesced reads of
// bbuf[j,:]). Lane 0 accumulates the gated coordinate update. Cross-wave
// reduction via LDS ds_add_f32 atomics.
//   sbuf[i,:]      = mean_j silu(a_i + b_j + dist*W1_d + b1)   (b1 folded in a)
//   coords_new[i]  = coords[i] + mean_j silu(c_i+d_j+dist*wcd) * rel_ij
// ---------------------------------------------------------------------------
__global__ void pair_kernel(const float* __restrict__ aa,     // x@W1_i + b1
                            const float* __restrict__ bbuf,   // x@W1_j
                            const float* __restrict__ cc,     // x@Wc_i + bc
                            const float* __restrict__ dd,     // x@Wc_j
                            const float* __restrict__ coords,
                            const float* __restrict__ W1,     // row 128 = W1_d
                            const float* __restrict__ Wc,     // Wc[128] = Wc_d
                            float* __restrict__ sbuf,
                            float* __restrict__ coords_out) {
    __shared__ float s_lds[DDIM];
    __shared__ float dlt[3];

    const int i    = blockIdx.x;
    const int tid  = threadIdx.x;
    const int lane = tid & 31;
    const int wave = tid >> 5;

    if (tid < DDIM) s_lds[tid] = 0.0f;
    if (tid < 3)    dlt[tid]   = 0.0f;
    __syncthreads();

    const float a0   = aa[i * DDIM + lane];
    const float a1   = aa[i * DDIM + lane + 32];
    const float w1d0 = W1[128 * DDIM + lane];
    const float w1d1 = W1[128 * DDIM + lane + 32];
    const float ci   = cc[i];
    const float wcd  = Wc[128];
    const float cx = coords[i * 3 + 0];
    const float cy = coords[i * 3 + 1];
    const float cz = coords[i * 3 + 2];

    float acc0 = 0.f, acc1 = 0.f;
    float dxa = 0.f, dya = 0.f, dza = 0.f;

    for (int j = wave; j < NPTS; j += 8) {
        const float b0  = bbuf[j * DDIM + lane];
        const float b1v = bbuf[j * DDIM + lane + 32];
        const float jx = coords[j * 3 + 0];
        const float jy = coords[j * 3 + 1];
        const float jz = coords[j * 3 + 2];
        const float rx = cx - jx, ry = cy - jy, rz = cz - jz;
        const float dist = sqrtf(rx * rx + ry * ry + rz * rz);

        acc0 += silu_fast(a0 + b0  + dist * w1d0);
        acc1 += silu_fast(a1 + b1v + dist * w1d1);

        const float wv = silu_fast(ci + dd[j] + dist * wcd);
        if (lane == 0) {
            dxa += wv * rx;
            dya += wv * ry;
            dza += wv * rz;
        }
    }

    atomicAdd(&s_lds[lane],      acc0);
    atomicAdd(&s_lds[lane + 32], acc1);
    if (lane == 0) {
        atomicAdd(&dlt[0], dxa);
        atomicAdd(&dlt[1], dya);
        atomicAdd(&dlt[2], dza);
    }
    __syncthreads();

    const float inv = 1.0f / (float)NPTS;
    if (tid < DDIM) sbuf[i * DDIM + tid] = s_lds[tid] * inv;
    if (tid < 3)    coords_out[i * 3 + tid] = coords[i * 3 + tid] + dlt[tid] * inv;
}

// ---------------------------------------------------------------------------
// Launch. Input order: x, coords, W1(129x64), b1(64), W2(64x64), b2(64),
// Wc(129x1), bc(1). Output: msg_x (1024*64) then coords_new (1024*3), f32.
// ---------------------------------------------------------------------------
extern "C" void kernel_launch(void* const* d_in, const int* in_sizes, int n_in,
                              void* d_out, int out_size, void* d_ws, size_t ws_size,
                              hipStream_t stream) {
    const float* x      = (const float*)d_in[0];
    const float* coords = (const float*)d_in[1];
    const float* W1     = (const float*)d_in[2];
    const float* b1     = (const float*)d_in[3];
    const float* W2     = (const float*)d_in[4];
    const float* b2     = (const float*)d_in[5];
    const float* Wc     = (const float*)d_in[6];
    const float* bc     = (const float*)d_in[7];

    float* msg        = (float*)d_out;               // 1024*64
    float* coords_new = (float*)d_out + NPTS * DDIM; // 1024*3

    float* ws   = (float*)d_ws;
    float* aa   = ws;                   // 1024*64
    float* bbuf = ws + 65536;           // 1024*64
    float* sbuf = ws + 131072;          // 1024*64
    float* cc   = ws + 196608;          // 1024
    float* dd   = ws + 197632;          // 1024

    // a = x @ W1_i + b1 ; b = x @ W1_j     (W1_j = rows 64..127 of W1)
    wmma_gemm_1024x64x64<<<NPTS / 16, 128, 0, stream>>>(x, W1,              b1,      aa);
    wmma_gemm_1024x64x64<<<NPTS / 16, 128, 0, stream>>>(x, W1 + DDIM * DDIM, nullptr, bbuf);
    gemv_wc<<<NPTS / 256, 256, 0, stream>>>(x, Wc, bc, cc, dd);

    pair_kernel<<<NPTS, 256, 0, stream>>>(aa, bbuf, cc, dd, coords, W1, Wc,
                                          sbuf, coords_new);

    // msg = mean_j silu(h) @ W2 + b2
    wmma_gemm_1024x64x64<<<NPTS / 16, 128, 0, stream>>>(sbuf, W2, b2, msg);
}